// DRACTransformChaserFruitbot_88837103550497
// MI455X (gfx1250) — compile-verified
//
#include <hip/hip_runtime.h>
#include <stdint.h>

#define PAD      3
#define HW       64
#define CH       3
#define PIX      (HW * HW)        // 4096
#define NELEM    (CH * PIX)       // 12288
#define NTHREADS 256
#define PER_TH   (NELEM / NTHREADS)   // 48

__device__ __forceinline__ int reflect64(int v) {
    // jnp reflect pad: -1 -> 1, -3 -> 3, 64 -> 62, 66 -> 60
    v = (v < 0) ? -v : v;
    v = (v > HW - 1) ? (2 * (HW - 1) - v) : v;
    return v;
}

__global__ __launch_bounds__(NTHREADS)
void aug_crop_jitter_kernel(const int*   __restrict__ x,
                            const int*   __restrict__ offs_h,
                            const int*   __restrict__ offs_w,
                            const float* __restrict__ jitter_b,
                            const float* __restrict__ jitter_c,
                            int*         __restrict__ out) {
    __shared__ int   tile[NELEM];      // 48 KB staged crop (raw int32 0..255)
    __shared__ float ch_sum[CH];

    const int b = blockIdx.x;
    const int t = threadIdx.x;

    const int   oh     = offs_h[b];
    const int   ow     = offs_w[b];
    const float delta  = (jitter_b[b] * 2.0f - 1.0f) * 0.05f;
    const float factor = 1.0f + (jitter_c[b] * 2.0f - 1.0f) * 0.05f;

    if (t < CH) ch_sum[t] = 0.0f;

    // ---- Stage the reflect-padded random crop into LDS via the CDNA5
    //      async copy engine (per-lane gather, ASYNCcnt-tracked). ----
    const long long gbase = (long long)(uintptr_t)x + 4ll * (long long)b * NELEM;
#pragma unroll
    for (int i = 0; i < PER_TH; ++i) {
        const int idx = t + NTHREADS * i;
        const int c   = i >> 4;                 // exact: 16 iters per channel
        const int pix = idx - c * PIX;
        const int r   = pix >> 6;
        const int col = pix & (HW - 1);
        const int sr  = reflect64(r + oh - PAD);
        const int sc  = reflect64(col + ow - PAD);
        const long long gaddr = gbase + 4ll * (c * PIX + sr * HW + sc);
        const unsigned  ldsoff = (unsigned)(uintptr_t)&tile[idx];
        asm volatile("global_load_async_to_lds_b32 %0, %1, off"
                     :: "v"(ldsoff), "v"(gaddr)
                     : "memory");
    }
    asm volatile("s_wait_asynccnt 0" ::: "memory");
    __syncthreads();

    // ---- Pass 1: per-channel spatial sum (raw int accumulation). ----
    float acc0 = 0.0f, acc1 = 0.0f, acc2 = 0.0f;
#pragma unroll
    for (int i = 0; i < 16; ++i)  acc0 += (float)tile[t + NTHREADS * i];
#pragma unroll
    for (int i = 16; i < 32; ++i) acc1 += (float)tile[t + NTHREADS * i];
#pragma unroll
    for (int i = 32; i < 48; ++i) acc2 += (float)tile[t + NTHREADS * i];

    atomicAdd(&ch_sum[0], acc0);   // ds_add_f32
    atomicAdd(&ch_sum[1], acc1);
    atomicAdd(&ch_sum[2], acc2);
    __syncthreads();

    const float s = 1.0f / 255.0f;
    float mean[CH];
#pragma unroll
    for (int c = 0; c < CH; ++c)
        mean[c] = ch_sum[c] * (s / (float)PIX);  // mean of crop (pre-brightness)

    // ---- Pass 2: brightness + contrast-about-mean + clip + requantize. ----
    // mean(x+delta) = mean(x)+delta  =>  (x01-mean_tot) = (v-mean_crop)
    const long long obase = (long long)b * NELEM;
#pragma unroll
    for (int i = 0; i < PER_TH; ++i) {
        const int   idx = t + NTHREADS * i;
        const int   c   = i >> 4;
        const float v   = (float)tile[idx] * s;
        const float m   = mean[c];
        float y = (v - m) * factor + m + delta;
        y = fminf(fmaxf(y, 0.0f), 1.0f);
        out[obase + idx] = (int)__builtin_rintf(y * 255.0f);  // v_rndne: half-to-even
    }
}

extern "C" void kernel_launch(void* const* d_in, const int* in_sizes, int n_in,
                              void* d_out, int out_size, void* d_ws, size_t ws_size,
                              hipStream_t stream) {
    const int*   x        = (const int*)d_in[0];
    const int*   offs_h   = (const int*)d_in[1];
    const int*   offs_w   = (const int*)d_in[2];
    const float* jitter_b = (const float*)d_in[3];
    const float* jitter_c = (const float*)d_in[4];
    int*         out      = (int*)d_out;

    const int B = in_sizes[1];   // 2048 samples
    aug_crop_jitter_kernel<<<B, NTHREADS, 0, stream>>>(
        x, offs_h, offs_w, jitter_b, jitter_c, out);
}